// CTCLoss_58961311040250
// MI455X (gfx1250) — compile-verified
//
#include <hip/hip_runtime.h>
#include <hip/hip_bf16.h>

// Problem constants (match the reference).
#define Tn 2000
#define Bn 64
#define Vn 163
#define Ln 200
#define Sn 401           // 2*L + 1
#define NTH 416          // 13 wave32s; >= Sn, multiple of 32
#define ROWPAD 176       // padded row buffer (>= Vn)
#define NEGF (-1e30f)
#define LOG2E_F 1.4426950408889634f
#define LN2_F 0.6931471805599453f

// Native base-2 transcendentals (v_exp_f32 / v_log_f32); fall back to libm.
#if __has_builtin(__builtin_amdgcn_exp2f)
#define EXP2F(x) __builtin_amdgcn_exp2f(x)
#else
#define EXP2F(x) exp2f(x)
#endif
#if __has_builtin(__builtin_amdgcn_logf)
#define LOG2F(x) __builtin_amdgcn_logf(x)
#else
#define LOG2F(x) log2f(x)
#endif

// ---------------------------------------------------------------------------
// CDNA5 async global->LDS copy of one f32 element per lane.
//   VDST vgpr  = per-lane LDS byte address
//   SADDR pair = uniform global base, VADDR = per-lane 32-bit byte offset
// Tracked by ASYNCcnt; consumer waits with s_wait_asynccnt.
// ---------------------------------------------------------------------------
__device__ __forceinline__ void async_f32_to_lds(const float* srcRow, float* dstLds, int v) {
    unsigned lds_addr = (unsigned)(unsigned long long)(const void*)(dstLds + v); // low 32b of shared aperture = LDS offset
    unsigned voff     = (unsigned)(v * 4);
    unsigned long long sbase = (unsigned long long)srcRow;
    asm volatile("global_load_async_to_lds_b32 %0, %1, %2"
                 :
                 : "v"(lds_addr), "v"(voff), "s"(sbase)
                 : "memory");
}

__device__ __forceinline__ void wait_async_lds() {
    asm volatile("s_wait_asynccnt 0" ::: "memory");
}

// ---------------------------------------------------------------------------
// Kernel 1: per-(t,b) row log2-sum-exp of logits. One wave32 per row.
// Output transposed: Z2T[b*Tn + t] (log2 domain) so the DP kernel preloads
// its Z-row with coalesced reads.
// ---------------------------------------------------------------------------
__global__ __launch_bounds__(256) void ctc_lse_kernel(const float* __restrict__ logits,
                                                      float* __restrict__ Z2T) {
    int gwave = (int)((blockIdx.x * blockDim.x + threadIdx.x) >> 5);
    int lane  = (int)(threadIdx.x & 31u);
    if (gwave >= Tn * Bn) return;

    const float* row = logits + (size_t)gwave * Vn;

    float y[6];
    float m = -INFINITY;
#pragma unroll
    for (int k = 0; k < 6; ++k) {
        int v = lane + 32 * k;
        y[k] = (v < Vn) ? row[v] * LOG2E_F : -INFINITY;
        m = fmaxf(m, y[k]);
    }
#pragma unroll
    for (int off = 16; off > 0; off >>= 1) m = fmaxf(m, __shfl_xor(m, off, 32));

    float s = 0.0f;
#pragma unroll
    for (int k = 0; k < 6; ++k) s += EXP2F(y[k] - m);   // -inf -> 0
#pragma unroll
    for (int off = 16; off > 0; off >>= 1) s += __shfl_xor(s, off, 32);

    if (lane == 0) {
        int t = gwave / Bn;
        int b = gwave % Bn;
        Z2T[(size_t)b * Tn + t] = m + LOG2F(s);         // log2(sum exp(x)), x pre-scaled by log2e
    }
}

// ---------------------------------------------------------------------------
// Kernel 2: CTC alpha recursion, one workgroup per utterance.
// One thread per extended-label state s in [0, 401). All math in log2 domain
// (native v_exp_f32 / v_log_f32 are base-2). Logits row for step t+1 streams
// into LDS via global_load_async_to_lds_b32 while step t computes.
// One workgroup barrier per time step (double-buffered alpha).
// Steps t >= act_len are identity in the reference (alpha frozen), so the
// loop runs only to Tlim = min(act_len, Tn) and the freeze-select is gone.
// ---------------------------------------------------------------------------
__global__ __launch_bounds__(NTH, 1) void ctc_alpha_kernel(const float* __restrict__ logits,
                                                           const int* __restrict__ labels,
                                                           const int* __restrict__ act_lens,
                                                           const int* __restrict__ label_lens,
                                                           const float* __restrict__ Z2T,
                                                           float* __restrict__ losses) {
    __shared__ float alpha[2][NTH];
    __shared__ float rowbuf[2][ROWPAD];
    __shared__ float zrow[Tn];
    __shared__ int   lab_sh[Ln];

    const int b   = (int)blockIdx.x;
    const int tid = (int)threadIdx.x;
    const int s   = tid;

    const int act_len = act_lens[b];
    const int lab_len = label_lens[b];
    const int Tlim    = (act_len < Tn) ? act_len : Tn;   // alpha is frozen beyond this

    // Preload labels and this utterance's Z2 row (coalesced) into LDS.
    for (int i = tid; i < Ln; i += NTH) lab_sh[i] = labels[b * Ln + i];
    for (int i = tid; i < Tlim; i += NTH) zrow[i] = Z2T[(size_t)b * Tn + i];

    const size_t rstride = (size_t)Bn * Vn;            // elements between time steps
    const float* base    = logits + (size_t)b * Vn;    // row (t=0, b)

    // Stage row t=0.
    if (tid < Vn) {
        async_f32_to_lds(base, &rowbuf[0][0], tid);
        wait_async_lds();
    }
    __syncthreads();   // rowbuf[0], zrow, lab_sh visible to all waves

    // Per-thread static state config.
    const bool valid = (s < 2 * lab_len + 1) && (s < Sn);
    int  extc   = 0;       // BLANK
    bool allow2 = false;   // may take alpha[s-2]
    if (s < Sn && (s & 1)) {
        int lab = lab_sh[(s - 1) >> 1];
        extc = lab;
        if (s >= 3) allow2 = (lab != lab_sh[(s - 3) >> 1]);
    }

    // alpha_0
    {
        float lp0 = rowbuf[0][extc] * LOG2E_F - zrow[0];
        alpha[0][tid] = ((s < 2) && valid) ? lp0 : NEGF;
    }

    // Stage row t=1 while others finish init.
    if (tid < Vn && Tlim > 1) async_f32_to_lds(base + rstride, &rowbuf[1][0], tid);

    for (int t = 1; t < Tlim; ++t) {
        const int cur = t & 1;
        const int prv = cur ^ 1;

        if (tid < Vn) wait_async_lds();   // row t landed in rowbuf[cur]
        __syncthreads();                  // publish rowbuf[cur] and alpha[prv]

        // Kick off the DMA for row t+1 into the buffer everyone is done reading.
        if (tid < Vn && (t + 1) < Tlim) {
            async_f32_to_lds(base + (size_t)(t + 1) * rstride, &rowbuf[prv][0], tid);
        }

        float a  = (s < Sn)            ? alpha[prv][s]     : NEGF;
        float a1 = (s >= 1 && s < Sn)  ? alpha[prv][s - 1] : NEGF;
        float a2 = allow2              ? alpha[prv][s - 2] : NEGF;

        float m  = fmaxf(a, fmaxf(a1, a2));
        float sm = EXP2F(a - m) + EXP2F(a1 - m) + EXP2F(a2 - m);
        float lp = rowbuf[cur][extc] * LOG2E_F - zrow[t];
        float nw = m + LOG2F(sm) + lp;

        alpha[cur][tid] = valid ? nw : NEGF;
    }

    __syncthreads();
    if (tid == 0) {
        const int fin = (Tlim - 1) & 1;
        int end = 2 * lab_len;
        float a_last = alpha[fin][end];
        float a_prev = (lab_len > 0) ? alpha[fin][end - 1] : NEGF;
        float m   = fmaxf(a_last, a_prev);
        float ll2 = m + LOG2F(EXP2F(a_last - m) + EXP2F(a_prev - m));
        losses[b] = -ll2 * LN2_F;         // back to natural-log domain
    }
}

// ---------------------------------------------------------------------------
// Kernel 3: total = sum(losses) / sum(act_lens)
// ---------------------------------------------------------------------------
__global__ __launch_bounds__(64) void ctc_reduce_kernel(const float* __restrict__ losses,
                                                        const int* __restrict__ act_lens,
                                                        float* __restrict__ out) {
    __shared__ float sl[Bn];
    __shared__ float sa[Bn];
    int i = (int)threadIdx.x;
    sl[i] = losses[i];
    sa[i] = (float)act_lens[i];
    __syncthreads();
    if (i == 0) {
        float S = 0.0f, A = 0.0f;
        for (int k = 0; k < Bn; ++k) { S += sl[k]; A += sa[k]; }
        out[0] = S / A;
    }
}

// ---------------------------------------------------------------------------
// Launcher. Inputs (setup_inputs order): logits f32[T*B*V], labels i32[B*L],
// act_lens i32[B], label_lens i32[B]. Output: 1 f32. Workspace: Z2T (B*T f32)
// + losses (B f32) = ~512.25 KB, fully overwritten every call.
// ---------------------------------------------------------------------------
extern "C" void kernel_launch(void* const* d_in, const int* in_sizes, int n_in,
                              void* d_out, int out_size, void* d_ws, size_t ws_size,
                              hipStream_t stream) {
    const float* logits     = (const float*)d_in[0];
    const int*   labels     = (const int*)d_in[1];
    const int*   act_lens   = (const int*)d_in[2];
    const int*   label_lens = (const int*)d_in[3];
    float*       out        = (float*)d_out;

    float* Z2T    = (float*)d_ws;                 // [Bn * Tn]
    float* losses = Z2T + (size_t)Bn * Tn;        // [Bn]

    // 1) log2-sum-exp over vocab for every (t, b) row: one wave32 per row.
    const int rows = Tn * Bn;                     // 128000
    const int wavesPerBlk = 256 / 32;
    const int nblk = (rows + wavesPerBlk - 1) / wavesPerBlk;  // 16000
    ctc_lse_kernel<<<nblk, 256, 0, stream>>>(logits, Z2T);

    // 2) sequential alpha recursion, one workgroup per utterance.
    ctc_alpha_kernel<<<Bn, NTH, 0, stream>>>(logits, labels, act_lens, label_lens, Z2T, losses);

    // 3) final scalar.
    ctc_reduce_kernel<<<1, 64, 0, stream>>>(losses, act_lens, out);
}